// TropicalConv2D_21655225107002
// MI455X (gfx1250) — compile-verified
//
#include <hip/hip_runtime.h>
#include <math.h>

// Tropical (max-plus) depthwise 3x3 conv, stride=1, pad=1, dil=1.
// x: (8,64,224,224) f32, kernel: (64,1,3,3) f32, out: (8,64,224,224) f32.
// Memory-bound (~206 MB @ 23.3 TB/s => ~9us floor). No WMMA mapping exists for
// max-plus; we use the gfx1250 async global->LDS DMA path for staging instead.

#define HH 224
#define WW 224
#define TILE_H 32
#define NROWS (TILE_H + 2)      // 34 staged rows (tile + halo)
#define LDSW 240                // padded LDS row stride (floats)
#define NCHUNK (WW / 4)         // 56 float4 chunks per row
#define NTHREADS 256

#if defined(__has_builtin)
#if __has_builtin(__builtin_amdgcn_global_load_async_to_lds_b128) && \
    __has_builtin(__builtin_amdgcn_s_wait_asynccnt)
#define USE_ASYNC_LDS 1
#endif
#endif

// Native vector types (HIP_vector_type is rejected by the builtins).
typedef int   v4i __attribute__((vector_size(16)));
typedef float v4f __attribute__((ext_vector_type(4)));
// Address-space-qualified pointee types for the async DMA builtin.
typedef __attribute__((address_space(1))) v4i gv4i;  // global
typedef __attribute__((address_space(3))) v4i lv4i;  // LDS

__global__ __launch_bounds__(NTHREADS)
void tropical_conv3x3_kernel(const float* __restrict__ x,
                             const float* __restrict__ kern,
                             float* __restrict__ out)
{
    __shared__ float lds[NROWS * LDSW];

    const int tid   = threadIdx.x;
    const int plane = blockIdx.y;          // b*64 + c
    const int chan  = plane & 63;
    const int row0  = blockIdx.x * TILE_H;

    const float* xp = x   + (size_t)plane * (HH * WW);
    float*       op = out + (size_t)plane * (HH * WW);

    // 9 per-channel taps: uniform across the block -> scalar loads.
    float w[9];
#pragma unroll
    for (int i = 0; i < 9; ++i) w[i] = kern[chan * 9 + i];

    const float NEG = -INFINITY;

    // Column halo: LDS index 3 holds x col -1, index 228 holds x col 224.
    for (int r = tid; r < NROWS; r += NTHREADS) {
        lds[r * LDSW + 3]   = NEG;
        lds[r * LDSW + 228] = NEG;
    }

    // Stage rows [row0-1, row0+TILE_H] into LDS. Data for x col c lives at
    // LDS index c+4, so 16-byte chunks land 16B-aligned.
    const int njobs = NROWS * NCHUNK;      // 34 * 56 = 1904
    for (int job = tid; job < njobs; job += NTHREADS) {
        const int lr = job / NCHUNK;
        const int ck = job - lr * NCHUNK;
        const int gr = row0 + lr - 1;
        float* dst = &lds[lr * LDSW + 4 + ck * 4];
        if (gr >= 0 && gr < HH) {
            const float* src = xp + (size_t)gr * WW + ck * 4;
#if defined(USE_ASYNC_LDS)
            __builtin_amdgcn_global_load_async_to_lds_b128(
                (gv4i*)src, (lv4i*)dst, 0, 0);
#else
            *(v4f*)dst = *(const v4f*)src;
#endif
        } else {
            v4f m = {NEG, NEG, NEG, NEG};
            *(v4f*)dst = m;
        }
    }

#if defined(USE_ASYNC_LDS)
    __builtin_amdgcn_s_wait_asynccnt(0);
#endif
    __syncthreads();

    // Compute: each thread produces 4 outputs per job; 32*56 = 1792 chunks.
    const int cjobs = TILE_H * NCHUNK;
    for (int job = tid; job < cjobs; job += NTHREADS) {
        const int r  = job / NCHUNK;       // 0..31 within tile
        const int cc = job - r * NCHUNK;
        const int c  = cc * 4;

        float a0, a1, a2, a3;
#pragma unroll
        for (int lr = 0; lr < 3; ++lr) {
            // LDS row r+lr holds x row (row0 + r + lr - 1); base idx = x col c-1.
            const float* rp = &lds[(r + lr) * LDSW + 3 + c];
            const float v0 = rp[0], v1 = rp[1], v2 = rp[2];
            const float v3 = rp[3], v4 = rp[4], v5 = rp[5];
            const float k0 = w[lr * 3 + 0], k1 = w[lr * 3 + 1], k2 = w[lr * 3 + 2];
            const float m0 = fmaxf(fmaxf(v0 + k0, v1 + k1), v2 + k2);
            const float m1 = fmaxf(fmaxf(v1 + k0, v2 + k1), v3 + k2);
            const float m2 = fmaxf(fmaxf(v2 + k0, v3 + k1), v4 + k2);
            const float m3 = fmaxf(fmaxf(v3 + k0, v4 + k1), v5 + k2);
            if (lr == 0) { a0 = m0; a1 = m1; a2 = m2; a3 = m3; }
            else {
                a0 = fmaxf(a0, m0); a1 = fmaxf(a1, m1);
                a2 = fmaxf(a2, m2); a3 = fmaxf(a3, m3);
            }
        }

        v4f o = {a0, a1, a2, a3};
        // Write-once output: nontemporal so the 192MB L2 keeps input resident.
        __builtin_nontemporal_store(o, (v4f*)(op + (size_t)(row0 + r) * WW + c));
    }
}

extern "C" void kernel_launch(void* const* d_in, const int* in_sizes, int n_in,
                              void* d_out, int out_size, void* d_ws, size_t ws_size,
                              hipStream_t stream) {
    const float* x = (const float*)d_in[0];
    const float* k = (const float*)d_in[1];
    float* out     = (float*)d_out;
    // stride/padding/dilation (d_in[2..4]) are compile-time 1/1/1 per setup.
    dim3 grid(HH / TILE_H, 8 * 64);   // (7, 512)
    tropical_conv3x3_kernel<<<grid, NTHREADS, 0, stream>>>(x, k, out);
}